// LightGCN_34376918237819
// MI455X (gfx1250) — compile-verified
//
#include <hip/hip_runtime.h>
#include <hip/hip_bf16.h>
#include <cstdint>
#include <cstddef>

#define NUM_USERS 200000
#define NUM_ITEMS 100000
#define NNODES    300000
#define EMB_DIM   64
#define NUM_EDGES 1250000

// ---------------------------------------------------------------------------
// CDNA5 async global->LDS path (guarded; falls back to sync loads if the
// builtin is absent so the file always compiles).
// ---------------------------------------------------------------------------
#if defined(__HIP_DEVICE_COMPILE__) && defined(__has_builtin)
#if __has_builtin(__builtin_amdgcn_global_load_async_to_lds_b32)
#define HAVE_ASYNC_LDS 1
#endif
#endif
#ifndef HAVE_ASYNC_LDS
#define HAVE_ASYNC_LDS 0
#endif

#if HAVE_ASYNC_LDS
typedef __attribute__((address_space(1))) int glob_int_t;
typedef __attribute__((address_space(3))) int lds_int_t;
__device__ __forceinline__ void async_copy_b32(const void* g, void* l) {
  __builtin_amdgcn_global_load_async_to_lds_b32((glob_int_t*)g, (lds_int_t*)l,
                                                /*offset=*/0, /*cpol=*/0);
}
#endif

// ---------------------------------------------------------------------------
// scatter: y[col] += norm * x[row].  One wave per edge (2 f32 per lane).
// Edge tuples are double-buffered in LDS via async DMA; each wave consumes
// exactly the LDS slots its own lanes loaded, so only per-wave
// s_wait_asynccnt ordering is needed (no block barriers).
// Placed FIRST in the file so the disasm snippet shows this (hot) kernel.
// ---------------------------------------------------------------------------
__global__ __launch_bounds__(256) void lgcn_scatter(
    const int* __restrict__ row, const int* __restrict__ col,
    const float* __restrict__ nrm, const float* __restrict__ x,
    float* __restrict__ y) {
  const int t = threadIdx.x;
  const int wv = t >> 5;
  const int ln = t & 31;
#if HAVE_ASYNC_LDS
  __shared__ int rowS[2][256];
  __shared__ int colS[2][256];
  __shared__ float nrmS[2][256];
  const int stride = (int)gridDim.x * 256;
  int base = (int)blockIdx.x * 256;
  if (base >= NUM_EDGES) return;
  {  // prefetch chunk 0 (indices clamped so issue count is uniform)
    int e = base + t;
    if (e > NUM_EDGES - 1) e = NUM_EDGES - 1;
    async_copy_b32(row + e, &rowS[0][t]);
    async_copy_b32(col + e, &colS[0][t]);
    async_copy_b32(nrm + e, &nrmS[0][t]);
  }
  int buf = 0;
  for (; base < NUM_EDGES; base += stride, buf ^= 1) {
    const int nbase = base + stride;
    if (nbase < NUM_EDGES) {  // prefetch next chunk, then wait on current
      int e = nbase + t;
      if (e > NUM_EDGES - 1) e = NUM_EDGES - 1;
      async_copy_b32(row + e, &rowS[buf ^ 1][t]);
      async_copy_b32(col + e, &colS[buf ^ 1][t]);
      async_copy_b32(nrm + e, &nrmS[buf ^ 1][t]);
      asm volatile("s_wait_asynccnt 3" ::: "memory");
    } else {
      asm volatile("s_wait_asynccnt 0" ::: "memory");
    }
    const int cnt = min(256, NUM_EDGES - base);
    const int jend = min(cnt, (wv + 1) * 32);
    for (int j = wv * 32; j < jend; ++j) {
      const int r = rowS[buf][j];
      const int c = colS[buf][j];
      const float nv = nrmS[buf][j];
      const float2 v = *(const float2*)(x + (size_t)r * EMB_DIM + ln * 2);
      float* dst = y + (size_t)c * EMB_DIM + ln * 2;
      unsafeAtomicAdd(dst, v.x * nv);
      unsafeAtomicAdd(dst + 1, v.y * nv);
    }
  }
#else
  (void)wv;
  const int gw = ((int)blockIdx.x * 256 + t) >> 5;
  const int nw = ((int)gridDim.x * 256) >> 5;
  for (int e = gw; e < NUM_EDGES; e += nw) {
    const int r = row[e];
    const int c = col[e];
    const float nv = nrm[e];
    const float2 v = *(const float2*)(x + (size_t)r * EMB_DIM + ln * 2);
    float* dst = y + (size_t)c * EMB_DIM + ln * 2;
    unsafeAtomicAdd(dst, v.x * nv);
    unsafeAtomicAdd(dst + 1, v.y * nv);
  }
#endif
}

// ---------------------------------------------------------------------------
// init: x = emb, y = 0, out[acc slots] = 0.25*emb, out[copy slots] = emb
// d_out layout: [users_final | user_emb | items_final | item_emb]
// ---------------------------------------------------------------------------
__global__ __launch_bounds__(256) void lgcn_init(
    const float* __restrict__ ue, const float* __restrict__ ie,
    float* __restrict__ x, float* __restrict__ y, float* __restrict__ out) {
  const size_t total4 = (size_t)NNODES * EMB_DIM / 4;
  const size_t ub4 = (size_t)NUM_USERS * EMB_DIM / 4;
  const size_t ib4 = (size_t)NUM_ITEMS * EMB_DIM / 4;
  for (size_t i = (size_t)blockIdx.x * 256 + threadIdx.x; i < total4;
       i += (size_t)gridDim.x * 256) {
    float4 v;
    size_t acc4, cpy4;
    if (i < ub4) {
      v = ((const float4*)ue)[i];
      acc4 = i;              // users_final region
      cpy4 = i + ub4;        // user_emb copy region
    } else {
      v = ((const float4*)ie)[i - ub4];
      acc4 = i + ub4;        // items_final region
      cpy4 = i + ub4 + ib4;  // item_emb copy region
    }
    ((float4*)x)[i] = v;
    ((float4*)y)[i] = make_float4(0.f, 0.f, 0.f, 0.f);
    ((float4*)out)[acc4] =
        make_float4(0.25f * v.x, 0.25f * v.y, 0.25f * v.z, 0.25f * v.w);
    ((float4*)out)[cpy4] = v;
  }
}

__global__ __launch_bounds__(256) void lgcn_zero_dis(float* __restrict__ dis) {
  const int n = blockIdx.x * 256 + threadIdx.x;
  if (n < NNODES) dis[n] = 0.f;
}

__global__ __launch_bounds__(256) void lgcn_deg(const int* __restrict__ col,
                                                float* __restrict__ dis) {
  const int e = blockIdx.x * 256 + threadIdx.x;
  if (e < NUM_EDGES) unsafeAtomicAdd(&dis[col[e]], 1.0f);
}

__global__ __launch_bounds__(256) void lgcn_invsqrt(float* __restrict__ dis) {
  const int n = blockIdx.x * 256 + threadIdx.x;
  if (n < NNODES) {
    const float d = dis[n];
    dis[n] = (d > 0.f) ? rsqrtf(d) : 0.f;
  }
}

__global__ __launch_bounds__(256) void lgcn_norm(const int* __restrict__ row,
                                                 const int* __restrict__ col,
                                                 const float* __restrict__ dis,
                                                 float* __restrict__ nrm) {
  const int e = blockIdx.x * 256 + threadIdx.x;
  if (e < NUM_EDGES) nrm[e] = dis[row[e]] * dis[col[e]];
}

// ---------------------------------------------------------------------------
// combine: acc += 0.25*y ; x = y ; y = 0
// ---------------------------------------------------------------------------
__global__ __launch_bounds__(256) void lgcn_combine(float* __restrict__ x,
                                                    float* __restrict__ y,
                                                    float* __restrict__ out) {
  const size_t total4 = (size_t)NNODES * EMB_DIM / 4;
  const size_t ub4 = (size_t)NUM_USERS * EMB_DIM / 4;
  for (size_t i = (size_t)blockIdx.x * 256 + threadIdx.x; i < total4;
       i += (size_t)gridDim.x * 256) {
    const float4 v = ((const float4*)y)[i];
    ((float4*)x)[i] = v;
    ((float4*)y)[i] = make_float4(0.f, 0.f, 0.f, 0.f);
    const size_t acc4 = (i < ub4) ? i : (i + ub4);
    float4 a = ((const float4*)out)[acc4];
    a.x += 0.25f * v.x;
    a.y += 0.25f * v.y;
    a.z += 0.25f * v.z;
    a.w += 0.25f * v.w;
    ((float4*)out)[acc4] = a;
  }
}

extern "C" void kernel_launch(void* const* d_in, const int* in_sizes, int n_in,
                              void* d_out, int out_size, void* d_ws,
                              size_t ws_size, hipStream_t stream) {
  const float* ue = (const float*)d_in[0];
  const float* ie = (const float*)d_in[1];
  const int* ei = (const int*)d_in[2];  // int32 (JAX x64 disabled)
  const int* row = ei;
  const int* col = ei + NUM_EDGES;
  float* out = (float*)d_out;

  // workspace: x | y | deg_inv_sqrt | norm  (~160 MB)
  float* x = (float*)d_ws;
  float* y = x + (size_t)NNODES * EMB_DIM;
  float* dis = y + (size_t)NNODES * EMB_DIM;
  float* nrm = dis + NNODES;

  const int g_elem = (NNODES * EMB_DIM / 4 + 255) / 256;  // 18750
  const int g_edge = (NUM_EDGES + 255) / 256;             // 4883
  const int g_node = (NNODES + 255) / 256;                // 1172

  lgcn_init<<<g_elem, 256, 0, stream>>>(ue, ie, x, y, out);
  lgcn_zero_dis<<<g_node, 256, 0, stream>>>(dis);
  lgcn_deg<<<g_edge, 256, 0, stream>>>(col, dis);
  lgcn_invsqrt<<<g_node, 256, 0, stream>>>(dis);
  lgcn_norm<<<g_edge, 256, 0, stream>>>(row, col, dis, nrm);

  for (int l = 0; l < 3; ++l) {
    lgcn_scatter<<<2048, 256, 0, stream>>>(row, col, nrm, x, y);
    lgcn_combine<<<g_elem, 256, 0, stream>>>(x, y, out);
  }

  (void)in_sizes;
  (void)n_in;
  (void)out_size;
  (void)ws_size;
}